// CComponents_17849884082373
// MI455X (gfx1250) — compile-verified
//
#include <hip/hip_runtime.h>

#define BATCH 256
#define HEADS 16
#define SIDE  14
#define PIX   196          // SIDE*SIDE
#define KMAX  49
#define NMAPS (BATCH * HEADS)
#define BIG   PIX          // background sentinel label

// ---- CDNA5 async global->LDS path (ASYNCcnt), guarded so compile can't fail ----
#if defined(__has_builtin)
#  if __has_builtin(__builtin_amdgcn_global_load_async_to_lds_b32) && \
      __has_builtin(__builtin_amdgcn_s_wait_asynccnt)
#    define USE_ASYNC_LDS 1
#  endif
#endif
#ifndef USE_ASYNC_LDS
#  define USE_ASYNC_LDS 0
#endif

typedef __attribute__((address_space(1))) void gvoid_as;   // global void
typedef __attribute__((address_space(3))) void lvoid_as;   // LDS void
typedef __attribute__((address_space(1))) int  gint_as;    // global int
typedef __attribute__((address_space(3))) int  lint_as;    // LDS int

__global__ __launch_bounds__(256)
void cc_blob_kernel(const float* __restrict__ x, float* __restrict__ out) {
  const int m = blockIdx.x;          // map id = b*HEADS + h
  const int t = threadIdx.x;         // 0..255 (8 wave32s)

  __shared__ float         xs[PIX];
  __shared__ int           lab[PIX];
  __shared__ unsigned char fgs[PIX];
  __shared__ int           scan[256];
  __shared__ int           comp[PIX];
  __shared__ int           changed;

  // ---------- Phase 1: stage input map into LDS (async DMA on gfx1250) ----------
#if USE_ASYNC_LDS
  if (t < PIX) {
    __builtin_amdgcn_global_load_async_to_lds_b32(
        (gint_as*)(gvoid_as*)(x + (size_t)m * PIX + t),
        (lint_as*)(lvoid_as*)&xs[t],
        /*offset=*/0, /*cpol=*/0);
  }
  __builtin_amdgcn_s_wait_asynccnt(0);
#else
  if (t < PIX) xs[t] = x[(size_t)m * PIX + t];
#endif
  if (t == 0) changed = 0;
  __syncthreads();

  // ---------- Phase 2: init labels ----------
  if (t < PIX) {
    const bool f = xs[t] > 0.0f;
    fgs[t] = f ? 1 : 0;
    lab[t] = f ? t : BIG;
  }
  __syncthreads();

  // ---------- Phase 3: min-label propagation + pointer jumping ----------
  // Invariant: lab[p] is the raster index of some pixel in p's component, <= p.
  // Fixed point = min raster index of the component (identical to reference).
  const int r = t / SIDE;
  const int c = t - r * SIDE;
  for (int iter = 0; iter < PIX; ++iter) {
    int nl = BIG;
    if (t < PIX && fgs[t]) {
      int mv = lab[t];
      const int r0 = (r > 0) ? r - 1 : 0;
      const int r1 = (r < SIDE - 1) ? r + 1 : SIDE - 1;
      const int c0 = (c > 0) ? c - 1 : 0;
      const int c1 = (c < SIDE - 1) ? c + 1 : SIDE - 1;
      for (int rr = r0; rr <= r1; ++rr) {
        const int base = rr * SIDE;
        for (int cc = c0; cc <= c1; ++cc) {
          const int v = lab[base + cc];
          mv = (v < mv) ? v : mv;
        }
      }
      // pointer-jump twice (accelerated convergence; preserves invariant)
      if (mv < BIG) mv = lab[mv];
      if (mv < BIG) mv = lab[mv];
      nl = mv;
    }
    __syncthreads();                          // all reads of lab done
    if (t < PIX && nl != lab[t]) { lab[t] = nl; changed = 1; }
    __syncthreads();                          // writes visible
    const int ch = changed;
    __syncthreads();                          // everyone sampled 'changed'
    if (t == 0) changed = 0;
    if (!ch) break;                           // uniform across the block
  }

  // ---------- Phase 4: representatives -> ranks via inclusive scan ----------
  scan[t] = (t < PIX && fgs[t] && lab[t] == t) ? 1 : 0;
  __syncthreads();
  for (int off = 1; off < 256; off <<= 1) {
    const int self = scan[t];
    const int add  = (t >= off) ? scan[t - off] : 0;
    __syncthreads();
    scan[t] = self + add;
    __syncthreads();
  }
  const int nb = scan[PIX - 1];               // number of components
  if (t < PIX) comp[t] = fgs[t] ? (scan[lab[t]] - 1) : -1;
  __syncthreads();

  // ---------- Phase 5: emit outputs (store-bandwidth bound) ----------
  const size_t NK   = (size_t)NMAPS * KMAX;                 // 200704
  float* obatch = out;
  float* ohead  = out + NK;
  float* omask  = out + 2 * NK;
  float* ovalid = out + 2 * NK + (size_t)NMAPS * KMAX * PIX;

  const int b = m / HEADS;
  const int h = m - b * HEADS;
  if (t < KMAX) {
    const size_t o = (size_t)m * KMAX + t;
    obatch[o] = (float)b;
    ohead[o]  = (float)h;
    ovalid[o] = (t < nb) ? 1.0f : 0.0f;
  }

  // masks: KMAX*PIX = 9604 floats per map; PIX % 4 == 0 so each float4 stays
  // inside one k-row. Consecutive threads -> consecutive float4 (b128) stores.
  float4* mb = (float4*)(omask + (size_t)m * (KMAX * PIX));
  for (int i = t; i < (KMAX * PIX) / 4; i += 256) {
    const int flat = i * 4;
    const int k  = flat / PIX;
    const int p0 = flat - k * PIX;
    float4 v;
    v.x = (comp[p0 + 0] == k) ? 1.0f : 0.0f;
    v.y = (comp[p0 + 1] == k) ? 1.0f : 0.0f;
    v.z = (comp[p0 + 2] == k) ? 1.0f : 0.0f;
    v.w = (comp[p0 + 3] == k) ? 1.0f : 0.0f;
    mb[i] = v;
  }
}

extern "C" void kernel_launch(void* const* d_in, const int* in_sizes, int n_in,
                              void* d_out, int out_size, void* d_ws, size_t ws_size,
                              hipStream_t stream) {
  (void)in_sizes; (void)n_in; (void)out_size; (void)d_ws; (void)ws_size;
  const float* x = (const float*)d_in[0];
  float* out = (float*)d_out;
  cc_blob_kernel<<<NMAPS, 256, 0, stream>>>(x, out);
}